// Div_15719580304337
// MI455X (gfx1250) — compile-verified
//
#include <hip/hip_runtime.h>
#include <stdint.h>

// TABLE_SCALE computed in f64 exactly as Python, then cast to f32 at use sites.
#define TSF ((float)(2.0 / 0.01 / 65535.0))

__device__ __forceinline__ float quantize_f(float y) {
    float r = rintf(y / TSF);                       // jnp.round == rne
    return fminf(fmaxf(r, -32768.0f), 32767.0f);    // clip to qint16
}

// Quantized reciprocal of ax (>=0), matching the reference's piecewise logic
// with f64-computed-then-f32-cast constants.
__device__ __forceinline__ float recip_piecewise(float ax) {
    // left line (LLINE = 1e-5 .. 0.01)
    const float Ly0  = (float)(1.0 / 1e-5);
    const float Lnum = (float)(1.0 / 0.01 - 1.0 / 1e-5);
    const float Lden = (float)(0.01 - 1e-5);
    // dense table [0.01, 1.0]: lookup step is f64->f32; linspace step is pure f32
    const float DK = (float)((1.0 - 0.01) / 255.0);
    const float DL = (1.0f - 0.01f) / 255.0f;
    // sparse table [1.0, 7.0]
    const float SK = (float)((7.0 - 1.0) / 255.0);
    const float SL = (7.0f - 1.0f) / 255.0f;
    // right line (RLINE = 7 .. 20)
    const float Ry0  = (float)(1.0 / 7.0);
    const float Rnum = (float)(1.0 / 20.0 - 1.0 / 7.0);
    const float Rden = (float)(20.0 - 7.0);

    if (ax < 1e-5f)  return quantize_f(Ly0);
    if (ax < 0.01f)  return quantize_f(Ly0 + ((ax - 1e-5f) * Lnum) / Lden);
    if (ax < 1.0f) {
        float idx = fminf(fmaxf(rintf((ax - 0.01f) / DK), 0.0f), 255.0f);
        float xs  = (idx == 255.0f) ? 1.0f : __fadd_rn(0.01f, __fmul_rn(idx, DL));
        return quantize_f(1.0f / xs);
    }
    if (ax < 7.0f) {
        float idx = fminf(fmaxf(rintf((ax - 1.0f) / SK), 0.0f), 255.0f);
        float xs  = (idx == 255.0f) ? 7.0f : __fadd_rn(1.0f, __fmul_rn(idx, SL));
        return quantize_f(1.0f / xs);
    }
    if (ax < 20.0f) return quantize_f(Ry0 + ((ax - 7.0f) * Rnum) / Rden);
    return quantize_f((float)(1.0 / 20.0));
}

// ---------------- Kernel A: build 65536-entry signed recip LUT --------------
__global__ __launch_bounds__(256) void build_table_kernel(
    const float* __restrict__ s2p, short* __restrict__ tab) {
    int t = blockIdx.x * blockDim.x + threadIdx.x;   // 0..65535
    if (t >= 65536) return;
    float s2 = *s2p;
    float x2 = (float)(t - 32768) * s2;
    float r  = recip_piecewise(fabsf(x2));
    if (!(x2 >= 0.0f)) r = -r;                       // sign = where(x2>=0, 1, -1)
    tab[t] = (short)(int)r;                          // exact, |r| <= 32768
}

// ---------------- element op (LUT in LDS) -----------------------------------
__device__ __forceinline__ float div_one(int q1v, int q2v, float s1, float osc,
                                         const short* tab) {
    float rf   = (float)tab[q2v + 32768];   // ds_load_i16 gather
    float bb   = rf * TSF;                  // recip_q * TABLE_SCALE
    float aa   = (float)q1v * s1;           // data1 * data1_scale
    float prod = aa * bb;
    float o    = rintf(prod / osc);         // exact IEEE divide + rne
    return fminf(fmaxf(o, -32768.0f), 32767.0f);
}

// ---------------- Kernel B: main streaming kernel ---------------------------
__global__ __launch_bounds__(256) void div_lut_kernel(
    const int4* __restrict__ q1, const int4* __restrict__ q2,
    const float* __restrict__ s1p, const float* __restrict__ osp,
    const short* __restrict__ gtab, float4* __restrict__ out, long long nvec) {
    __shared__ short tab[65536];            // 128 KB of the 320 KB WGP LDS

#if __has_builtin(__builtin_amdgcn_global_load_async_to_lds_b128)
    {   // CDNA5 async copy path: global -> LDS without touching VGPRs
        typedef int v4i __attribute__((vector_size(16)));
        typedef __attribute__((address_space(1))) v4i* gvp_t;
        typedef __attribute__((address_space(3))) v4i* lvp_t;
        typedef const __attribute__((address_space(1))) char* gp_t;
        typedef __attribute__((address_space(3))) char*       lp_t;
        gp_t g = (gp_t)(const char*)gtab;
        lp_t l = (lp_t)(char*)tab;
        for (int j = (int)threadIdx.x * 16; j < 131072; j += (int)blockDim.x * 16) {
            __builtin_amdgcn_global_load_async_to_lds_b128(
                (gvp_t)(g + j), (lvp_t)(l + j), 0, 0);
        }
        asm volatile("s_wait_asynccnt 0" ::: "memory");
    }
#else
    {   // fallback: load->store copy
        const int4* src = (const int4*)gtab;
        int4*       dst = (int4*)tab;
        for (int j = threadIdx.x; j < 8192; j += blockDim.x) dst[j] = src[j];
    }
#endif
    __syncthreads();

    float s1  = *s1p;
    float osc = *osp;
    long long stride = (long long)gridDim.x * blockDim.x;
    for (long long i = (long long)blockIdx.x * blockDim.x + threadIdx.x;
         i < nvec; i += stride) {
        long long nx = i + stride;          // prefetch next chunk (global_prefetch_b8)
        if (nx < nvec) { __builtin_prefetch(&q1[nx]); __builtin_prefetch(&q2[nx]); }
        int4 a = q1[i];
        int4 b = q2[i];
        float4 o;
        o.x = div_one(a.x, b.x, s1, osc, tab);
        o.y = div_one(a.y, b.y, s1, osc, tab);
        o.z = div_one(a.z, b.z, s1, osc, tab);
        o.w = div_one(a.w, b.w, s1, osc, tab);
        out[i] = o;
    }
}

// ---------------- fallback / tail: fully direct computation -----------------
__global__ __launch_bounds__(256) void div_direct_kernel(
    const int* __restrict__ q1, const int* __restrict__ q2,
    const float* __restrict__ s1p, const float* __restrict__ s2p,
    const float* __restrict__ osp, float* __restrict__ out,
    long long start, long long n) {
    float s1 = *s1p, s2 = *s2p, osc = *osp;
    long long stride = (long long)gridDim.x * blockDim.x;
    for (long long i = start + (long long)blockIdx.x * blockDim.x + threadIdx.x;
         i < n; i += stride) {
        float x2 = (float)q2[i] * s2;
        float r  = recip_piecewise(fabsf(x2));
        if (!(x2 >= 0.0f)) r = -r;
        float prod = ((float)q1[i] * s1) * (r * TSF);
        float o = rintf(prod / osc);
        out[i] = fminf(fmaxf(o, -32768.0f), 32767.0f);
    }
}

extern "C" void kernel_launch(void* const* d_in, const int* in_sizes, int n_in,
                              void* d_out, int out_size, void* d_ws, size_t ws_size,
                              hipStream_t stream) {
    const int*   q1 = (const int*)d_in[0];
    const int*   q2 = (const int*)d_in[1];
    const float* s1 = (const float*)d_in[2];
    const float* s2 = (const float*)d_in[3];
    const float* os = (const float*)d_in[4];
    float* out = (float*)d_out;
    long long N = (long long)out_size;
    if (N <= 0) return;

    bool ws_ok = (ws_size >= 65536 * sizeof(short)) &&
                 ((((uintptr_t)d_ws) & 0xF) == 0) &&
                 ((((uintptr_t)q1) & 0xF) == 0) && ((((uintptr_t)q2) & 0xF) == 0) &&
                 ((((uintptr_t)out) & 0xF) == 0);

    if (ws_ok && N >= 4) {
        short* tab = (short*)d_ws;
        build_table_kernel<<<256, 256, 0, stream>>>(s2, tab);
        long long nvec = N / 4;
        long long b = (nvec + 255) / 256;
        int blocks = (int)(b > 4096 ? 4096 : b);
        div_lut_kernel<<<blocks, 256, 0, stream>>>(
            (const int4*)q1, (const int4*)q2, s1, os, tab, (float4*)out, nvec);
        long long tail_start = nvec * 4;
        if (tail_start < N)
            div_direct_kernel<<<1, 64, 0, stream>>>(q1, q2, s1, s2, os, out,
                                                    tail_start, N);
    } else {
        long long b = (N + 255) / 256;
        int blocks = (int)(b > 8192 ? 8192 : b);
        if (blocks < 1) blocks = 1;
        div_direct_kernel<<<blocks, 256, 0, stream>>>(q1, q2, s1, s2, os, out, 0, N);
    }
}